// PointNet2_18889266167881
// MI455X (gfx1250) — compile-verified
//
#include <hip/hip_runtime.h>
#include <hip/hip_bf16.h>

typedef __attribute__((ext_vector_type(2))) float v2f;
typedef __attribute__((ext_vector_type(8))) float v8f;

// ---------------------------------------------------------------------------
// Layout transforms
// ---------------------------------------------------------------------------
__global__ void __launch_bounds__(256)
pn2_transpose_c2p(const float* __restrict__ cm, float* __restrict__ pm, int B, int N) {
  int tid = blockIdx.x * blockDim.x + threadIdx.x;
  if (tid >= B * N) return;
  int b = tid / N, n = tid % N;
  for (int d = 0; d < 3; ++d)
    pm[((size_t)b * N + n) * 3 + d] = cm[((size_t)b * 3 + d) * N + n];
}

__global__ void __launch_bounds__(256)
pn2_gather_xyz(const float* __restrict__ xyz, const int* __restrict__ cidx,
               float* __restrict__ nxyz, int B, int N, int M) {
  int tid = blockIdx.x * blockDim.x + threadIdx.x;
  if (tid >= B * M) return;
  int b = tid / M, m = tid % M;
  int gi = cidx[(size_t)b * M + m];
  const float* p = xyz + ((size_t)b * N + gi) * 3;
  float* q = nxyz + ((size_t)b * M + m) * 3;
  q[0] = p[0]; q[1] = p[1]; q[2] = p[2];
}

// ---------------------------------------------------------------------------
// Farthest point sampling: one block per batch, xyz + dists resident in LDS.
// Matches jnp.argmax tie-break (lowest index of the max).
// ---------------------------------------------------------------------------
#define FPS_T 256
__global__ void __launch_bounds__(FPS_T)
pn2_fps(const float* __restrict__ xyz, int* __restrict__ out, int N, int M) {
  __shared__ float sx[2048], sy[2048], sz[2048], sd[2048];
  __shared__ float rv[FPS_T];
  __shared__ int   ri[FPS_T];
  __shared__ int   s_last;
  int b = blockIdx.x, t = threadIdx.x;
  const float* xb = xyz + (size_t)b * N * 3;
  for (int n = t; n < N; n += FPS_T) {
    sx[n] = xb[(size_t)n * 3 + 0];
    sy[n] = xb[(size_t)n * 3 + 1];
    sz[n] = xb[(size_t)n * 3 + 2];
    sd[n] = 1e10f;
  }
  if (t == 0) { out[(size_t)b * M] = 0; s_last = 0; }
  __syncthreads();
  for (int it = 1; it < M; ++it) {
    int last = s_last;
    float px = sx[last], py = sy[last], pz = sz[last];
    float bestv = -1.0f; int besti = 0;
    for (int n = t; n < N; n += FPS_T) {
      float dx = sx[n] - px, dy = sy[n] - py, dz = sz[n] - pz;
      float d = dx * dx + dy * dy + dz * dz;
      float dn = sd[n];
      dn = d < dn ? d : dn;
      sd[n] = dn;
      if (dn > bestv) { bestv = dn; besti = n; }
    }
    rv[t] = bestv; ri[t] = besti;
    __syncthreads();
    for (int s = FPS_T / 2; s > 0; s >>= 1) {
      if (t < s) {
        float v2 = rv[t + s]; int i2 = ri[t + s];
        if (v2 > rv[t] || (v2 == rv[t] && i2 < ri[t])) { rv[t] = v2; ri[t] = i2; }
      }
      __syncthreads();
    }
    if (t == 0) { s_last = ri[0]; out[(size_t)b * M + it] = ri[0]; }
    __syncthreads();
  }
}

// ---------------------------------------------------------------------------
// Ball query: one wave32 per center; ballot/popcount prefix emits first K
// in-radius indices in ascending order (== sort(where(...))[:K]), pad = first.
// ---------------------------------------------------------------------------
__global__ void __launch_bounds__(128)
pn2_ball_query(const float* __restrict__ xyz, const float* __restrict__ nxyz,
               int* __restrict__ gidx, int B, int N, int M, int K, float r2) {
  int wave = blockIdx.x * (blockDim.x >> 5) + (threadIdx.x >> 5);
  int lane = threadIdx.x & 31;
  if (wave >= B * M) return;
  int b = wave / M, m = wave % M;
  const float* c = nxyz + ((size_t)b * M + m) * 3;
  float cx = c[0], cy = c[1], cz = c[2];
  const float* xb = xyz + (size_t)b * N * 3;
  int* g = gidx + ((size_t)b * M + m) * K;
  int cnt = 0, first = -1;
  for (int base = 0; base < N && cnt < K; base += 32) {
    int n = base + lane;
    bool pred = false;
    if (n < N) {
      float dx = xb[(size_t)n * 3 + 0] - cx;
      float dy = xb[(size_t)n * 3 + 1] - cy;
      float dz = xb[(size_t)n * 3 + 2] - cz;
      pred = (dx * dx + dy * dy + dz * dz) <= r2;
    }
    unsigned mask = (unsigned)__ballot((int)pred);
    if (first < 0 && mask) first = base + __builtin_ctz(mask);
    int rank = __builtin_popcount(mask & ((1u << lane) - 1u));
    if (pred && cnt + rank < K) g[cnt + rank] = n;
    cnt += __builtin_popcount(mask);
  }
  if (cnt > K) cnt = K;
  if (first < 0) first = 0;
  for (int j = cnt + lane; j < K; j += 32) g[j] = first;
}

// ---------------------------------------------------------------------------
// Build grouped GEMM operand X (B, Cf+3, M*K): features first, then rel-xyz.
// ---------------------------------------------------------------------------
__global__ void __launch_bounds__(256)
pn2_build_group(const float* __restrict__ feat, const float* __restrict__ xyz,
                const float* __restrict__ nxyz, const int* __restrict__ gidx,
                float* __restrict__ X, int B, int Cf, int Np, int M, int K) {
  size_t tid = (size_t)blockIdx.x * blockDim.x + threadIdx.x;
  size_t tot = (size_t)B * M * K;
  if (tid >= tot) return;
  int k = (int)(tid % K);
  int m = (int)((tid / K) % M);
  int b = (int)(tid / ((size_t)K * M));
  int gi = gidx[((size_t)b * M + m) * K + k];
  int Nn = M * K, Ct = Cf + 3, n = m * K + k;
  float* Xb = X + (size_t)b * Ct * Nn;
  const float* fb = feat + (size_t)b * Cf * Np;
  for (int c = 0; c < Cf; ++c)
    Xb[(size_t)c * Nn + n] = fb[(size_t)c * Np + gi];
  const float* p  = xyz  + ((size_t)b * Np + gi) * 3;
  const float* cc = nxyz + ((size_t)b * M  + m)  * 3;
  Xb[(size_t)(Cf + 0) * Nn + n] = p[0] - cc[0];
  Xb[(size_t)(Cf + 1) * Nn + n] = p[1] - cc[1];
  Xb[(size_t)(Cf + 2) * Nn + n] = p[2] - cc[2];
}

// ---------------------------------------------------------------------------
// FP32 WMMA GEMM: Y[co,n] = epilogue(sum_c W[co,c]*X[c,n]).
// One wave computes a 32(co) x 128(n) tile: 2 A streams x 8 N-subtiles,
// 16x V_WMMA_F32_16X16X4_F32 per K-step, explicit 2-stage software pipeline
// (next K-step's A/B fragments load while current WMMAs execute).
// mode=1: relu(g*(acc+b)+bt)   mode=0: acc+b
// ---------------------------------------------------------------------------
__global__ void __launch_bounds__(32)
pn2_gemm(const float* __restrict__ W, const float* __restrict__ X,
         float* __restrict__ Y, const float* __restrict__ bias,
         const float* __restrict__ gam, const float* __restrict__ bet,
         int co, int cin, int Nn, int mode) {
  const int lane = threadIdx.x;
  const int half = lane >> 4;        // K sub-pair select (0: K0/1, 1: K2/3)
  const int l16  = lane & 15;
  const int n0 = blockIdx.x * 128;
  const int m0 = blockIdx.y * 32;
  const int b  = blockIdx.z;
  const float* Xb = X + (size_t)b * cin * Nn;
  float* Yb = Y + (size_t)b * co * Nn;
  int mA = m0 + l16;       if (mA >= co) mA = co - 1;   // clamped row loads,
  int mB = m0 + 16 + l16;  if (mB >= co) mB = co - 1;   // stores masked later

  const size_t rs = (size_t)Nn;      // X row stride (floats)
  const size_t ks = 4 * rs;          // K-step stride
  const float* pA0 = W + (size_t)mA * cin + half * 2;
  const float* pA1 = W + (size_t)mB * cin + half * 2;
  const float* pB  = Xb + (size_t)(half * 2) * rs + n0 + l16;

  v8f acc[16] = {};
  const int cin4 = cin & ~3;
  const int nk = cin4 >> 2;

  if (nk > 0) {
    v2f a0, a1, bb[8];
    a0.x = pA0[0]; a0.y = pA0[1];
    a1.x = pA1[0]; a1.y = pA1[1];
    #pragma unroll
    for (int s = 0; s < 8; ++s) { bb[s].x = pB[s * 16]; bb[s].y = pB[rs + s * 16]; }

    for (int i = 1; i < nk; ++i) {
      pA0 += 4; pA1 += 4; pB += ks;
      v2f na0, na1, nb[8];                       // issue next K-step's loads
      na0.x = pA0[0]; na0.y = pA0[1];
      na1.x = pA1[0]; na1.y = pA1[1];
      #pragma unroll
      for (int s = 0; s < 8; ++s) { nb[s].x = pB[s * 16]; nb[s].y = pB[rs + s * 16]; }
      #pragma unroll
      for (int s = 0; s < 8; ++s)                // consume current fragments
        acc[s] = __builtin_amdgcn_wmma_f32_16x16x4_f32(false, a0, false, bb[s],
                                                       (short)0, acc[s], false, false);
      #pragma unroll
      for (int s = 0; s < 8; ++s)
        acc[8 + s] = __builtin_amdgcn_wmma_f32_16x16x4_f32(false, a1, false, bb[s],
                                                           (short)0, acc[8 + s], false, false);
      a0 = na0; a1 = na1;
      #pragma unroll
      for (int s = 0; s < 8; ++s) bb[s] = nb[s];
    }
    #pragma unroll
    for (int s = 0; s < 8; ++s)
      acc[s] = __builtin_amdgcn_wmma_f32_16x16x4_f32(false, a0, false, bb[s],
                                                     (short)0, acc[s], false, false);
    #pragma unroll
    for (int s = 0; s < 8; ++s)
      acc[8 + s] = __builtin_amdgcn_wmma_f32_16x16x4_f32(false, a1, false, bb[s],
                                                         (short)0, acc[8 + s], false, false);
  }

  if (cin4 < cin) {                  // tail: clamp addresses, select A to zero
    const int c0 = cin4 + half * 2;
    const int c1 = c0 + 1;
    const int c0c = c0 < cin ? c0 : cin - 1;
    const int c1c = c1 < cin ? c1 : cin - 1;
    float a0x = W[(size_t)mA * cin + c0c], a0y = W[(size_t)mA * cin + c1c];
    float a1x = W[(size_t)mB * cin + c0c], a1y = W[(size_t)mB * cin + c1c];
    v2f a0, a1;
    a0.x = (c0 < cin) ? a0x : 0.0f;  // zero A column => B contents irrelevant
    a0.y = (c1 < cin) ? a0y : 0.0f;
    a1.x = (c0 < cin) ? a1x : 0.0f;
    a1.y = (c1 < cin) ? a1y : 0.0f;
    const float* r0 = Xb + (size_t)c0c * rs + n0 + l16;
    const float* r1 = Xb + (size_t)c1c * rs + n0 + l16;
    #pragma unroll
    for (int s = 0; s < 8; ++s) {
      v2f bb; bb.x = r0[s * 16]; bb.y = r1[s * 16];
      acc[s]     = __builtin_amdgcn_wmma_f32_16x16x4_f32(false, a0, false, bb,
                                                         (short)0, acc[s], false, false);
      acc[8 + s] = __builtin_amdgcn_wmma_f32_16x16x4_f32(false, a1, false, bb,
                                                         (short)0, acc[8 + s], false, false);
    }
  }

  #pragma unroll
  for (int t = 0; t < 2; ++t) {
    #pragma unroll
    for (int r = 0; r < 8; ++r) {
      int mm = m0 + t * 16 + r + half * 8;   // C/D layout: VGPR r = rows r / r+8
      if (mm < co) {
        float bb = bias[mm];
        float gg = 1.0f, tt = 0.0f;
        if (mode) { gg = gam[mm]; tt = bet[mm]; }
        float* yr = Yb + (size_t)mm * rs + n0 + l16;
        #pragma unroll
        for (int s = 0; s < 8; ++s) {
          float v = acc[t * 8 + s][r] + bb;
          if (mode) { v = v * gg + tt; v = v > 0.0f ? v : 0.0f; }
          yr[s * 16] = v;
        }
      }
    }
  }
}

// ---------------------------------------------------------------------------
// Max over last (K) axis: out[b,c,m] = max_k in[b,c,m*K+k]
// ---------------------------------------------------------------------------
__global__ void __launch_bounds__(256)
pn2_maxpool_k(const float* __restrict__ in, float* __restrict__ out,
              int B, int C, int M, int K) {
  size_t tid = (size_t)blockIdx.x * blockDim.x + threadIdx.x;
  size_t tot = (size_t)B * C * M;
  if (tid >= tot) return;
  int m = (int)(tid % M);
  int c = (int)((tid / M) % C);
  int b = (int)(tid / ((size_t)M * C));
  const float* p = in + ((size_t)b * C + c) * ((size_t)M * K) + (size_t)m * K;
  float mx = -1e30f;
  for (int k = 0; k < K; ++k) mx = fmaxf(mx, p[k]);
  out[tid] = mx;
}

// ---------------------------------------------------------------------------
// Concat / broadcast helpers (dst is (B, Ct, N) channel-major)
// ---------------------------------------------------------------------------
__global__ void __launch_bounds__(256)
pn2_concat_copy(const float* __restrict__ src, float* __restrict__ dst,
                int B, int C, int N, int Ct, int coff) {
  size_t tid = (size_t)blockIdx.x * blockDim.x + threadIdx.x;
  size_t tot = (size_t)B * C * N;
  if (tid >= tot) return;
  int n = (int)(tid % N);
  int c = (int)((tid / N) % C);
  int b = (int)(tid / ((size_t)N * C));
  dst[((size_t)b * Ct + coff + c) * N + n] = src[((size_t)b * C + c) * N + n];
}

__global__ void __launch_bounds__(256)
pn2_broadcast_ch(const float* __restrict__ src /*(B,C)*/, float* __restrict__ dst,
                 int B, int C, int N, int Ct, int coff) {
  size_t tid = (size_t)blockIdx.x * blockDim.x + threadIdx.x;
  size_t tot = (size_t)B * C * N;
  if (tid >= tot) return;
  int n = (int)(tid % N);
  int c = (int)((tid / N) % C);
  int b = (int)(tid / ((size_t)N * C));
  dst[((size_t)b * Ct + coff + c) * N + n] = src[(size_t)b * C + c];
}

__global__ void __launch_bounds__(256)
pn2_concat_xyz_cm(const float* __restrict__ pm /*(B,M,3)*/, float* __restrict__ dst,
                  int B, int M, int Ct, int coff) {
  int tid = blockIdx.x * blockDim.x + threadIdx.x;
  if (tid >= B * M) return;
  int b = tid / M, m = tid % M;
  for (int d = 0; d < 3; ++d)
    dst[((size_t)b * Ct + coff + d) * M + m] = pm[((size_t)b * M + m) * 3 + d];
}

__global__ void __launch_bounds__(256)
pn2_build_feat0(const int* __restrict__ cls, const float* __restrict__ features,
                const float* __restrict__ points, float* __restrict__ dst,
                int B, int N, int Ct) {
  int tid = blockIdx.x * blockDim.x + threadIdx.x;
  if (tid >= B * N) return;
  int b = tid / N, n = tid % N;
  int cid = cls[b];
  float* db = dst + (size_t)b * Ct * N;
  for (int c = 0; c < 16; ++c) db[(size_t)c * N + n] = (c == cid) ? 1.0f : 0.0f;
  for (int d = 0; d < 3; ++d)
    db[(size_t)(16 + d) * N + n] = features[((size_t)b * 3 + d) * N + n];
  for (int d = 0; d < 3; ++d)
    db[(size_t)(19 + d) * N + n] = points[((size_t)b * 3 + d) * N + n];
}

// ---------------------------------------------------------------------------
// 3-NN (stable top-3 of smallest d2) + inverse-distance weights
// ---------------------------------------------------------------------------
__global__ void __launch_bounds__(256)
pn2_knn3(const float* __restrict__ fine /*(B,N,3)*/, const float* __restrict__ coarse /*(B,S,3)*/,
         int* __restrict__ idx3, float* __restrict__ w3, int B, int N, int S) {
  int tid = blockIdx.x * blockDim.x + threadIdx.x;
  if (tid >= B * N) return;
  int b = tid / N, n = tid % N;
  const float* p = fine + ((size_t)b * N + n) * 3;
  float px = p[0], py = p[1], pz = p[2];
  const float* cb = coarse + (size_t)b * S * 3;
  float d0 = 1e30f, d1 = 1e30f, d2 = 1e30f;
  int i0 = 0, i1 = 0, i2 = 0;
  for (int s = 0; s < S; ++s) {
    float dx = cb[(size_t)s * 3 + 0] - px;
    float dy = cb[(size_t)s * 3 + 1] - py;
    float dz = cb[(size_t)s * 3 + 2] - pz;
    float d = dx * dx + dy * dy + dz * dz;
    if (d < d0)      { d2 = d1; i2 = i1; d1 = d0; i1 = i0; d0 = d; i0 = s; }
    else if (d < d1) { d2 = d1; i2 = i1; d1 = d; i1 = s; }
    else if (d < d2) { d2 = d; i2 = s; }
  }
  float w0 = 1.0f / (d0 + 1e-8f);
  float w1 = 1.0f / (d1 + 1e-8f);
  float w2 = 1.0f / (d2 + 1e-8f);
  float ws = w0 + w1 + w2;
  size_t o = ((size_t)b * N + n) * 3;
  idx3[o + 0] = i0; idx3[o + 1] = i1; idx3[o + 2] = i2;
  w3[o + 0] = w0 / ws; w3[o + 1] = w1 / ws; w3[o + 2] = w2 / ws;
}

__global__ void __launch_bounds__(256)
pn2_interp_write(const float* __restrict__ fc /*(B,C,S)*/, const int* __restrict__ idx3,
                 const float* __restrict__ w3, float* __restrict__ dst /*(B,Ct,N)*/,
                 int B, int C, int S, int N, int Ct, int coff) {
  size_t tid = (size_t)blockIdx.x * blockDim.x + threadIdx.x;
  size_t tot = (size_t)B * C * N;
  if (tid >= tot) return;
  int n = (int)(tid % N);
  int c = (int)((tid / N) % C);
  int b = (int)(tid / ((size_t)N * C));
  const float* f = fc + ((size_t)b * C + c) * S;
  size_t o = ((size_t)b * N + n) * 3;
  float v = w3[o + 0] * f[idx3[o + 0]] +
            w3[o + 1] * f[idx3[o + 1]] +
            w3[o + 2] * f[idx3[o + 2]];
  dst[((size_t)b * Ct + coff + c) * N + n] = v;
}

// ---------------------------------------------------------------------------
// Host orchestration
// ---------------------------------------------------------------------------
static inline int cdiv(long long a, long long b) { return (int)((a + b - 1) / b); }

extern "C" void kernel_launch(void* const* d_in, const int* in_sizes, int n_in,
                              void* d_out, int out_size, void* d_ws, size_t ws_size,
                              hipStream_t stream) {
  (void)in_sizes; (void)n_in; (void)out_size; (void)ws_size;
  const int B = 8, N0 = 2048;
  const float* points   = (const float*)d_in[0];
  const float* features = (const float*)d_in[1];
  const int*   class_ids = (const int*)d_in[2];

  // ---- workspace bump allocator ----
  size_t off = 0;
  auto alloc = [&](size_t elems) -> void* {
    void* p = (char*)d_ws + off;
    off += ((elems * 4 + 255) & ~(size_t)255);
    return p;
  };
  float* xyz0 = (float*)alloc((size_t)B * 2048 * 3);
  int*   cidx1 = (int*)alloc((size_t)B * 512);
  float* nx1  = (float*)alloc((size_t)B * 512 * 3);
  int*   gidx1 = (int*)alloc((size_t)B * 512 * 64);
  float* f1   = (float*)alloc((size_t)B * 128 * 512);
  int*   cidx2 = (int*)alloc((size_t)B * 128);
  float* nx2  = (float*)alloc((size_t)B * 128 * 3);
  int*   gidx2 = (int*)alloc((size_t)B * 128 * 64);
  float* f2   = (float*)alloc((size_t)B * 256 * 128);
  float* f3   = (float*)alloc((size_t)B * 1024);
  float* fp1o = (float*)alloc((size_t)B * 256 * 128);
  int*   kidx = (int*)alloc((size_t)B * 2048 * 3);
  float* kw   = (float*)alloc((size_t)B * 2048 * 3);
  float* f1p  = (float*)alloc((size_t)B * 128 * 512);
  float* bufA = (float*)alloc((size_t)B * 128 * 32768);
  float* bufB = (float*)alloc((size_t)B * 128 * 32768);

  auto gemm_l = [&](int base, const float* X, float* Y, int co, int cin, int Nn) {
    dim3 g(Nn / 128, (co + 31) / 32, B);
    pn2_gemm<<<g, 32, 0, stream>>>((const float*)d_in[base], X, Y,
        (const float*)d_in[base + 1], (const float*)d_in[base + 2],
        (const float*)d_in[base + 3], co, cin, Nn, 1);
  };

  // ---- SA1: N=2048 -> M=512, K=64, r=0.2, 6 -> 64 -> 64 -> 128 ----
  pn2_transpose_c2p<<<cdiv(B * N0, 256), 256, 0, stream>>>(points, xyz0, B, N0);
  pn2_fps<<<B, FPS_T, 0, stream>>>(xyz0, cidx1, N0, 512);
  pn2_gather_xyz<<<cdiv(B * 512, 256), 256, 0, stream>>>(xyz0, cidx1, nx1, B, N0, 512);
  pn2_ball_query<<<cdiv(B * 512, 4), 128, 0, stream>>>(xyz0, nx1, gidx1, B, N0, 512, 64, 0.04f);
  pn2_build_group<<<cdiv((long long)B * 512 * 64, 256), 256, 0, stream>>>(
      features, xyz0, nx1, gidx1, bufA, B, 3, N0, 512, 64);
  gemm_l(3,  bufA, bufB, 64, 6, 32768);
  gemm_l(7,  bufB, bufA, 64, 64, 32768);
  gemm_l(11, bufA, bufB, 128, 64, 32768);
  pn2_maxpool_k<<<cdiv((long long)B * 128 * 512, 256), 256, 0, stream>>>(bufB, f1, B, 128, 512, 64);

  // ---- SA2: 512 -> 128, K=64, r=0.4, 131 -> 128 -> 128 -> 256 ----
  pn2_fps<<<B, FPS_T, 0, stream>>>(nx1, cidx2, 512, 128);
  pn2_gather_xyz<<<cdiv(B * 128, 256), 256, 0, stream>>>(nx1, cidx2, nx2, B, 512, 128);
  pn2_ball_query<<<cdiv(B * 128, 4), 128, 0, stream>>>(nx1, nx2, gidx2, B, 512, 128, 64, 0.16f);
  pn2_build_group<<<cdiv((long long)B * 128 * 64, 256), 256, 0, stream>>>(
      f1, nx1, nx2, gidx2, bufA, B, 128, 512, 128, 64);
  gemm_l(15, bufA, bufB, 128, 131, 8192);
  gemm_l(19, bufB, bufA, 128, 128, 8192);
  gemm_l(23, bufA, bufB, 256, 128, 8192);
  pn2_maxpool_k<<<cdiv((long long)B * 256 * 128, 256), 256, 0, stream>>>(bufB, f2, B, 256, 128, 64);

  // ---- GSA: concat(f2, p2) 259 -> 256 -> 512 -> 1024, global max ----
  pn2_concat_copy<<<cdiv((long long)B * 256 * 128, 256), 256, 0, stream>>>(f2, bufA, B, 256, 128, 259, 0);
  pn2_concat_xyz_cm<<<cdiv(B * 128, 256), 256, 0, stream>>>(nx2, bufA, B, 128, 259, 256);
  gemm_l(27, bufA, bufB, 256, 259, 128);
  gemm_l(31, bufB, bufA, 512, 256, 128);
  gemm_l(35, bufA, bufB, 1024, 512, 128);
  pn2_maxpool_k<<<cdiv((long long)B * 1024, 256), 256, 0, stream>>>(bufB, f3, B, 1024, 1, 128);

  // ---- FP1: broadcast f3, concat f2 -> 1280 -> 256 -> 256 ----
  pn2_concat_copy<<<cdiv((long long)B * 256 * 128, 256), 256, 0, stream>>>(f2, bufA, B, 256, 128, 1280, 0);
  pn2_broadcast_ch<<<cdiv((long long)B * 1024 * 128, 256), 256, 0, stream>>>(f3, bufA, B, 1024, 128, 1280, 256);
  gemm_l(39, bufA, bufB, 256, 1280, 128);
  gemm_l(43, bufB, fp1o, 256, 256, 128);

  // ---- FP2: 3-NN(p1 <- p2), concat(f1, interp) 384 -> 256 -> 128 ----
  pn2_knn3<<<cdiv(B * 512, 256), 256, 0, stream>>>(nx1, nx2, kidx, kw, B, 512, 128);
  pn2_concat_copy<<<cdiv((long long)B * 128 * 512, 256), 256, 0, stream>>>(f1, bufA, B, 128, 512, 384, 0);
  pn2_interp_write<<<cdiv((long long)B * 256 * 512, 256), 256, 0, stream>>>(
      fp1o, kidx, kw, bufA, B, 256, 128, 512, 384, 128);
  gemm_l(47, bufA, bufB, 256, 384, 512);
  gemm_l(51, bufB, f1p, 128, 256, 512);

  // ---- FP3: 3-NN(points <- p1), concat(onehot|feat|pts, interp) 150 -> 128 x3 ----
  pn2_knn3<<<cdiv(B * 2048, 256), 256, 0, stream>>>(xyz0, nx1, kidx, kw, B, 2048, 512);
  pn2_build_feat0<<<cdiv(B * 2048, 256), 256, 0, stream>>>(class_ids, features, points, bufA, B, 2048, 150);
  pn2_interp_write<<<cdiv((long long)B * 128 * 2048, 256), 256, 0, stream>>>(
      f1p, kidx, kw, bufA, B, 128, 512, 2048, 150, 22);
  gemm_l(55, bufA, bufB, 128, 150, 2048);
  gemm_l(59, bufB, bufA, 128, 128, 2048);
  gemm_l(63, bufA, bufB, 128, 128, 2048);   // f0 in bufB

  // ---- classifier head: cls_block (bn+relu) then final 50x128 (bias only) ----
  gemm_l(67, bufB, bufA, 128, 128, 2048);
  dim3 gf(2048 / 128, (50 + 31) / 32, B);
  pn2_gemm<<<gf, 32, 0, stream>>>((const float*)d_in[71], bufA, (float*)d_out,
      (const float*)d_in[72], (const float*)d_in[72], (const float*)d_in[72],
      50, 128, 2048, 0);
}